// SelectAction_64862596104471
// MI455X (gfx1250) — compile-verified
//
#include <hip/hip_runtime.h>

// Epsilon-greedy action selection for MI455X (gfx1250, wave32).
// B=131072 rows x 512 fp32 Q-values. Pure streaming kernel:
//   256 MB read once -> ~11 us floor at 23.3 TB/s.
// One wave32 per row; 128-bit non-temporal loads (read-once stream, avoid
// polluting the 192 MB L2); wave32 butterfly argmax via shuffles.

#define EG_BATCH      131072
#define EG_NACT       512
#define EG_EPS        0.05f
#define EG_WAVES_PER_BLOCK 8   // 256 threads / 32 lanes

typedef float v4f __attribute__((ext_vector_type(4)));

__global__ __launch_bounds__(256)
void SelectAction_epsgreedy_kernel(const float* __restrict__ q,
                                   const float* __restrict__ rand_u,
                                   const int*   __restrict__ rand_actions,
                                   int*         __restrict__ out)
{
    const int lane = threadIdx.x & 31;
    const int wave = threadIdx.x >> 5;
    const int row  = blockIdx.x * EG_WAVES_PER_BLOCK + wave;
    if (row >= EG_BATCH) return;

    const v4f* qrow = (const v4f*)(q + (size_t)row * EG_NACT);

    float best_v = -__builtin_inff();
    int   best_i = 0;

    // 512 floats = 128 float4s = 4 iterations x 32 lanes.
    // Iteration i: lanes 0..31 read one contiguous 512 B segment (coalesced).
    // Per-lane indices are strictly increasing across iterations, so strict '>'
    // preserves first-occurrence argmax semantics within a lane.
#pragma unroll
    for (int it = 0; it < 4; ++it) {
        const int vec  = it * 32 + lane;          // float4 index within row
        const v4f v    = __builtin_nontemporal_load(qrow + vec);
        const int base = vec * 4;
#pragma unroll
        for (int c = 0; c < 4; ++c) {
            const float f = v[c];
            if (f > best_v) { best_v = f; best_i = base + c; }
        }
    }

    // Wave32 butterfly reduction (ds_swizzle/permlane path).
    // Tie-break toward the smaller index to match jnp.argmax exactly.
#pragma unroll
    for (int off = 16; off > 0; off >>= 1) {
        const float ov = __shfl_xor(best_v, off, 32);
        const int   oi = __shfl_xor(best_i, off, 32);
        if (ov > best_v || (ov == best_v && oi < best_i)) {
            best_v = ov;
            best_i = oi;
        }
    }

    if (lane == 0) {
        const float u = rand_u[row];
        const int   a = (u < EG_EPS) ? rand_actions[row] : best_i;
        __builtin_nontemporal_store(a, out + row);
    }
}

extern "C" void kernel_launch(void* const* d_in, const int* in_sizes, int n_in,
                              void* d_out, int out_size, void* d_ws, size_t ws_size,
                              hipStream_t stream)
{
    (void)in_sizes; (void)n_in; (void)out_size; (void)d_ws; (void)ws_size;

    const float* q            = (const float*)d_in[0];  // [B, 512] fp32
    const float* rand_u       = (const float*)d_in[1];  // [B] fp32
    const int*   rand_actions = (const int*)  d_in[2];  // [B] int32
    int*         out          = (int*)d_out;            // [B] int32

    const int blocks = EG_BATCH / EG_WAVES_PER_BLOCK;   // 16384
    SelectAction_epsgreedy_kernel<<<blocks, 256, 0, stream>>>(q, rand_u, rand_actions, out);
}